// MOCOv2_300647711389
// MI455X (gfx1250) — compile-verified
//
#include <hip/hip_runtime.h>
#include <hip/hip_bf16.h>
#include <math.h>

typedef __attribute__((ext_vector_type(2))) float v2f;
typedef __attribute__((ext_vector_type(8))) float v8f;

#define W_IMG 608.0f
#define NB 32
#define NC 256
#define FDIM 12544   // 256*49
#define HID 1024
#define KQ 1024
#define INV_TEMP 100.0f

#define LDS_STRIDE 34          // even (8B-aligned b64) and 34*i mod 64 distinct for 16 rows
#define SPLITK_BIG 28          // 392 slabs of 32 -> 14 per split
#define SPLITK_SMALL 8         // 32 slabs of 32  -> 4 per split

// ---------------------------------------------------------------------------
// 1) Routed ROI-align: pick pyramid level from box size, 7x7 pool with 2x2
//    bilinear samples per cell. One block per batch element, thread = channel.
// ---------------------------------------------------------------------------
__global__ __launch_bounds__(256)
void roi_route_kernel(const float* __restrict__ f0, const float* __restrict__ f1,
                      const float* __restrict__ f2, const float* __restrict__ f3,
                      const float* __restrict__ bbox, float* __restrict__ out)
{
    __shared__ float s_fr[28];
    __shared__ int   s_lo[28], s_hi[28], s_v[28];

    const int b   = blockIdx.x;
    const int tid = threadIdx.x;

    const float x1b = bbox[b * 4 + 0], y1b = bbox[b * 4 + 1];
    const float x2b = bbox[b * 4 + 2], y2b = bbox[b * 4 + 3];
    const float bw = x2b - x1b, bh = y2b - y1b;
    float val = roundf(log2f(sqrtf(bw * bh) / W_IMG) + 4.0f);
    val = fminf(fmaxf(val, 2.0f), 5.0f);
    const int lvl = (int)val - 2;

    const float* feat; int S;
    switch (lvl) {
        case 0:  feat = f0; S = 152; break;
        case 1:  feat = f1; S = 76;  break;
        case 2:  feat = f2; S = 38;  break;
        default: feat = f3; S = 19;  break;
    }
    const float scale = (float)S / W_IMG;
    const float rx1 = x1b * scale, ry1 = y1b * scale;
    const float rw = fmaxf(x2b * scale - rx1, 1.0f);
    const float rh = fmaxf(y2b * scale - ry1, 1.0f);

    if (tid < 28) {
        const int axis = tid / 14, i = tid % 14;      // 0..13 = y axis, 14..27 = x axis
        const float g  = ((float)i + 0.5f) * 0.5f;    // (i+0.5)/sr
        const float c  = (axis == 0) ? (ry1 + g * (rh / 7.0f))
                                     : (rx1 + g * (rw / 7.0f));
        const float fs = (float)S;
        const int   valid = (c > -1.0f) && (c < fs);
        const float cc = fmaxf(c, 0.0f);
        const float fl = floorf(cc);
        const int   lo = (int)fminf(fl, fs - 1.0f);
        const int   hi = (lo + 1 < S - 1) ? (lo + 1) : (S - 1);
        const float fr = (fl >= fs - 1.0f) ? 0.0f : (cc - fl);
        s_lo[tid] = lo; s_hi[tid] = hi; s_fr[tid] = fr; s_v[tid] = valid;
    }
    __syncthreads();

    const int ch = tid;                              // 256 threads == 256 channels
    const float* fb = feat + (((size_t)b * NC + ch) * (size_t)(S * S));
    float* ob = out + (size_t)b * FDIM + (size_t)ch * 49;

    for (int cell = 0; cell < 49; ++cell) {
        const int oy = cell / 7, ox = cell % 7;
        float acc = 0.0f;
        #pragma unroll
        for (int sy = 0; sy < 2; ++sy) {
            #pragma unroll
            for (int sx = 0; sx < 2; ++sx) {
                const int j = oy * 2 + sy, i = ox * 2 + sx;
                if (s_v[j] && s_v[14 + i]) {
                    const int   yl = s_lo[j],      yh = s_hi[j];
                    const int   xl = s_lo[14 + i], xh = s_hi[14 + i];
                    const float fy = s_fr[j],      fx = s_fr[14 + i];
                    const float v00 = fb[yl * S + xl], v01 = fb[yl * S + xh];
                    const float v10 = fb[yh * S + xl], v11 = fb[yh * S + xh];
                    acc += (1.0f - fy) * ((1.0f - fx) * v00 + fx * v01)
                         +        fy  * ((1.0f - fx) * v10 + fx * v11);
                }
            }
        }
        ob[cell] = acc * 0.25f;
    }
}

// ---------------------------------------------------------------------------
// 2) WMMA fp32 GEMM with split-K:
//    partial[split, 32, N] = A[32, kRange] @ W[N, kRange]^T
//    V_WMMA_F32_16X16X4_F32; block = 8 waves, each wave one 16x16 D tile;
//    block tile = 32(M) x 64(N); A and B k-slabs staged in LDS (padded,
//    conflict-free, coalesced b128 staging loads).
// ---------------------------------------------------------------------------
__global__ __launch_bounds__(256)
void gemm32_wmma_kernel(const float* __restrict__ A, const float* __restrict__ W,
                        float* __restrict__ partial, int N, int K, int slabsPerSplit)
{
    __shared__ float ldsA[32 * LDS_STRIDE];
    __shared__ float ldsB[64 * LDS_STRIDE];

    const int tid   = threadIdx.x;
    const int wave  = tid >> 5;
    const int lane  = tid & 31;
    const int half  = lane >> 4;          // 0: lanes 0-15, 1: lanes 16-31
    const int l16   = lane & 15;
    const int mTile = wave & 1;           // 2 M tiles
    const int nTile = wave >> 1;          // 4 N tiles per block
    const int koff  = half * 2;           // ISA A/B layout: halves hold K{0,1}/K{2,3}
    const int aRow  = mTile * 16 + l16;
    const int bRow  = nTile * 16 + l16;
    const int nBase = blockIdx.x * 64;
    const int nCol  = nBase + bRow;

    // staging assignments
    const int arow_st = tid >> 3;                 // A: 32 rows x 32 cols, float4/thread
    const int acol_st = (tid & 7) * 4;
    const int brow_st = tid >> 2;                 // B: 64 rows x 32 cols, 2 float4/thread
    const int bcol_st = (tid & 3) * 8;

    const int kStart = blockIdx.y * slabsPerSplit * 32;
    const int kEnd   = min(kStart + slabsPerSplit * 32, K);

    v8f acc = {};

    for (int k0 = kStart; k0 < kEnd; k0 += 32) {
        // ---- stage A[0:32, k0:k0+32] (coalesced b128) ----
        {
            const float4 v = *reinterpret_cast<const float4*>(
                A + (size_t)arow_st * K + k0 + acol_st);
            float* dst = &ldsA[arow_st * LDS_STRIDE + acol_st];
            *reinterpret_cast<float2*>(dst + 0) = make_float2(v.x, v.y);
            *reinterpret_cast<float2*>(dst + 2) = make_float2(v.z, v.w);
        }
        // ---- stage W[nBase:nBase+64, k0:k0+32] (coalesced b128 x2) ----
        {
            const float* src = W + (size_t)(nBase + brow_st) * K + k0 + bcol_st;
            const float4 v0 = *reinterpret_cast<const float4*>(src);
            const float4 v1 = *reinterpret_cast<const float4*>(src + 4);
            float* dst = &ldsB[brow_st * LDS_STRIDE + bcol_st];
            *reinterpret_cast<float2*>(dst + 0) = make_float2(v0.x, v0.y);
            *reinterpret_cast<float2*>(dst + 2) = make_float2(v0.z, v0.w);
            *reinterpret_cast<float2*>(dst + 4) = make_float2(v1.x, v1.y);
            *reinterpret_cast<float2*>(dst + 6) = make_float2(v1.z, v1.w);
            // prefetch next weight slab (global_prefetch_b8)
            if (k0 + 32 < kEnd) __builtin_prefetch(src + 32, 0, 1);
        }
        __syncthreads();

        #pragma unroll
        for (int kk = 0; kk < 32; kk += 4) {
            v2f a, bf;
            const float2 av = *reinterpret_cast<const float2*>(
                &ldsA[aRow * LDS_STRIDE + kk + koff]);
            const float2 bv = *reinterpret_cast<const float2*>(
                &ldsB[bRow * LDS_STRIDE + kk + koff]);
            a.x = av.x;  a.y = av.y;       // A 16x4: halves hold K{0,1}/K{2,3}
            bf.x = bv.x; bf.y = bv.y;      // B 4x16: B[k][n] = W[n][k]
            acc = __builtin_amdgcn_wmma_f32_16x16x4_f32(
                false, a, false, bf, (short)0, acc, false, false);
        }
        __syncthreads();
    }

    #pragma unroll
    for (int r = 0; r < 8; ++r) {
        const int m = mTile * 16 + r + half * 8;   // C/D layout: VGPR r -> M=r / r+8
        partial[((size_t)blockIdx.y * NB + m) * N + nCol] = acc[r];
    }
}

// ---------------------------------------------------------------------------
// 2b) Deterministic fixed-order split-K reduction (+ optional bias).
// ---------------------------------------------------------------------------
__global__ __launch_bounds__(256)
void reduce_bias_kernel(const float* __restrict__ partial, const float* __restrict__ bias,
                        float* __restrict__ out, int N, int splits)
{
    const int i = blockIdx.x * blockDim.x + threadIdx.x;
    if (i >= NB * N) return;
    const int col = i % N;
    float s = 0.0f;
    for (int sp = 0; sp < splits; ++sp) s += partial[(size_t)sp * NB * N + i];
    out[i] = s + (bias ? bias[col] : 0.0f);
}

// ---------------------------------------------------------------------------
// 3) BatchNorm over batch axis (32 rows) + ReLU, in place. thread = column.
// ---------------------------------------------------------------------------
__global__ __launch_bounds__(256)
void bn_relu_kernel(float* __restrict__ h, const float* __restrict__ g,
                    const float* __restrict__ be, int N)
{
    const int col = blockIdx.x * blockDim.x + threadIdx.x;
    if (col >= N) return;
    float s = 0.0f, s2 = 0.0f;
    for (int b = 0; b < NB; ++b) {
        const float v = h[(size_t)b * N + col];
        s += v; s2 += v * v;
    }
    const float mu  = s * (1.0f / NB);
    const float var = s2 * (1.0f / NB) - mu * mu;
    const float inv = rsqrtf(var + 1e-5f);
    const float gg = g[col], bb = be[col];
    for (int b = 0; b < NB; ++b) {
        float v = (h[(size_t)b * N + col] - mu) * inv * gg + bb;
        h[(size_t)b * N + col] = v > 0.0f ? v : 0.0f;
    }
}

// ---------------------------------------------------------------------------
// 4) Row L2-normalize, in place. block = row.
// ---------------------------------------------------------------------------
__global__ __launch_bounds__(256)
void rownorm_kernel(float* __restrict__ x, int N)
{
    __shared__ float red[256];
    const int b = blockIdx.x, tid = threadIdx.x;
    float s = 0.0f;
    for (int j = tid; j < N; j += 256) {
        const float v = x[(size_t)b * N + j];
        s += v * v;
    }
    red[tid] = s; __syncthreads();
    for (int off = 128; off > 0; off >>= 1) {
        if (tid < off) red[tid] += red[tid + off];
        __syncthreads();
    }
    const float inv = 1.0f / fmaxf(sqrtf(red[0]), 1e-12f);
    for (int j = tid; j < N; j += 256) x[(size_t)b * N + j] *= inv;
}

// ---------------------------------------------------------------------------
// 5) l_pos: per-row dot(q,k). block = row.
// ---------------------------------------------------------------------------
__global__ __launch_bounds__(256)
void dotrow_kernel(const float* __restrict__ q, const float* __restrict__ k,
                   float* __restrict__ lpos, int N)
{
    __shared__ float red[256];
    const int b = blockIdx.x, tid = threadIdx.x;
    float s = 0.0f;
    for (int j = tid; j < N; j += 256)
        s += q[(size_t)b * N + j] * k[(size_t)b * N + j];
    red[tid] = s; __syncthreads();
    for (int off = 128; off > 0; off >>= 1) {
        if (tid < off) red[tid] += red[tid + off];
        __syncthreads();
    }
    if (tid == 0) lpos[b] = red[0];
}

// ---------------------------------------------------------------------------
// 6) Assemble logits (concat pos|neg, /TEMP) + int32 zeros (bit == 0.0f).
// ---------------------------------------------------------------------------
__global__ __launch_bounds__(256)
void assemble_kernel(const float* __restrict__ lpos, const float* __restrict__ lneg,
                     float* __restrict__ out)
{
    const int i  = blockIdx.x * blockDim.x + threadIdx.x;
    const int NL = NB * (KQ + 1);
    if (i < NL) {
        const int b = i / (KQ + 1), j = i % (KQ + 1);
        out[i] = (j == 0 ? lpos[b] : lneg[(size_t)b * KQ + (j - 1)]) * INV_TEMP;
    } else if (i < NL + NB) {
        out[i] = 0.0f;
    }
}

// ---------------------------------------------------------------------------
extern "C" void kernel_launch(void* const* d_in, const int* in_sizes, int n_in,
                              void* d_out, int out_size, void* d_ws, size_t ws_size,
                              hipStream_t stream)
{
    const float* fq0 = (const float*)d_in[0];
    const float* fq1 = (const float*)d_in[1];
    const float* fq2 = (const float*)d_in[2];
    const float* fq3 = (const float*)d_in[3];
    const float* fk0 = (const float*)d_in[4];
    const float* fk1 = (const float*)d_in[5];
    const float* fk2 = (const float*)d_in[6];
    const float* fk3 = (const float*)d_in[7];
    const float* q_bbox = (const float*)d_in[8];
    const float* k_bbox = (const float*)d_in[9];
    const float* w1_q = (const float*)d_in[10];
    const float* g_q  = (const float*)d_in[11];
    const float* be_q = (const float*)d_in[12];
    const float* w2_q = (const float*)d_in[13];
    const float* b2_q = (const float*)d_in[14];
    const float* w1_k = (const float*)d_in[15];
    const float* g_k  = (const float*)d_in[16];
    const float* be_k = (const float*)d_in[17];
    const float* w2_k = (const float*)d_in[18];
    const float* b2_k = (const float*)d_in[19];
    const float* queue = (const float*)d_in[20];
    float* out = (float*)d_out;

    // workspace layout (floats)
    float* ws = (float*)d_ws;
    float* alignedQ = ws;                              // 32*12544
    float* alignedK = alignedQ + (size_t)NB * FDIM;    // 32*12544
    float* hQ   = alignedK + (size_t)NB * FDIM;        // 32*1024
    float* hK   = hQ   + (size_t)NB * HID;
    float* qbuf = hK   + (size_t)NB * HID;
    float* kbuf = qbuf + (size_t)NB * HID;
    float* lneg = kbuf + (size_t)NB * HID;
    float* lpos = lneg + (size_t)NB * KQ;              // 32
    float* part = lpos + NB;                           // SPLITK_BIG*32*1024

    const int nElem = NB * HID;
    const dim3 blk(256);

    // 1) routed ROI-align (q and k paths)
    roi_route_kernel<<<NB, 256, 0, stream>>>(fq0, fq1, fq2, fq3, q_bbox, alignedQ);
    roi_route_kernel<<<NB, 256, 0, stream>>>(fk0, fk1, fk2, fk3, k_bbox, alignedK);

    // 2) layer 1 GEMMs (WMMA, split-K=28): [32,12544] @ [1024,12544]^T
    {
        const dim3 grid(HID / 64, SPLITK_BIG);
        const int slabs = (FDIM / 32) / SPLITK_BIG;    // 392/28 = 14
        gemm32_wmma_kernel<<<grid, blk, 0, stream>>>(alignedQ, w1_q, part, HID, FDIM, slabs);
        reduce_bias_kernel<<<(nElem + 255) / 256, blk, 0, stream>>>(part, nullptr, hQ, HID, SPLITK_BIG);
        gemm32_wmma_kernel<<<grid, blk, 0, stream>>>(alignedK, w1_k, part, HID, FDIM, slabs);
        reduce_bias_kernel<<<(nElem + 255) / 256, blk, 0, stream>>>(part, nullptr, hK, HID, SPLITK_BIG);
    }

    // 3) batch-norm + relu
    bn_relu_kernel<<<HID / 256, blk, 0, stream>>>(hQ, g_q, be_q, HID);
    bn_relu_kernel<<<HID / 256, blk, 0, stream>>>(hK, g_k, be_k, HID);

    // 4) layer 2 GEMMs (WMMA, split-K=8, fused bias in reduce)
    {
        const dim3 grid(HID / 64, SPLITK_SMALL);
        const int slabs = (HID / 32) / SPLITK_SMALL;   // 32/8 = 4
        gemm32_wmma_kernel<<<grid, blk, 0, stream>>>(hQ, w2_q, part, HID, HID, slabs);
        reduce_bias_kernel<<<(nElem + 255) / 256, blk, 0, stream>>>(part, b2_q, qbuf, HID, SPLITK_SMALL);
        gemm32_wmma_kernel<<<grid, blk, 0, stream>>>(hK, w2_k, part, HID, HID, slabs);
        reduce_bias_kernel<<<(nElem + 255) / 256, blk, 0, stream>>>(part, b2_k, kbuf, HID, SPLITK_SMALL);
    }

    // 5) row normalize
    rownorm_kernel<<<NB, 256, 0, stream>>>(qbuf, HID);
    rownorm_kernel<<<NB, 256, 0, stream>>>(kbuf, HID);

    // 6) l_pos and l_neg (WMMA, split-K=8): q @ queue^T
    dotrow_kernel<<<NB, 256, 0, stream>>>(qbuf, kbuf, lpos, HID);
    {
        const dim3 grid(KQ / 64, SPLITK_SMALL);
        const int slabs = (HID / 32) / SPLITK_SMALL;
        gemm32_wmma_kernel<<<grid, blk, 0, stream>>>(qbuf, queue, part, KQ, HID, slabs);
        reduce_bias_kernel<<<(nElem + 255) / 256, blk, 0, stream>>>(part, nullptr, lneg, KQ, SPLITK_SMALL);
    }

    // 7) assemble logits + zeros
    const int total = NB * (KQ + 1) + NB;
    assemble_kernel<<<(total + 255) / 256, blk, 0, stream>>>(lpos, lneg, out);
}